// InterpersonalGNN_70463233458394
// MI455X (gfx1250) — compile-verified
//
#include <hip/hip_runtime.h>

typedef __attribute__((ext_vector_type(2))) float v2f;
typedef __attribute__((ext_vector_type(8))) float v8f;

#define HID 128

// ---------------------------------------------------------------- degree prep
__global__ __launch_bounds__(256) void k_deg_init(float* __restrict__ deg, int n) {
  int i = blockIdx.x * 256 + threadIdx.x;
  if (i < n) deg[i] = 1.0f;                       // self-loop contribution
}

__global__ __launch_bounds__(256) void k_deg_edges(float* __restrict__ deg,
                                                   const int* __restrict__ dst, int e) {
  int i = blockIdx.x * 256 + threadIdx.x;
  if (i < e) atomicAdd(&deg[dst[i]], 1.0f);
}

__global__ __launch_bounds__(256) void k_rsqrt(float* __restrict__ deg, int n) {
  int i = blockIdx.x * 256 + threadIdx.x;
  if (i < n) deg[i] = __frsqrt_rn(deg[i]);        // deg >= 1 always (self-loops)
}

// ---------------------------------------------------------------- fp32 WMMA GEMM
// H[M x 128] = X[M x 128] * W[128 x 128], fp32 via V_WMMA_F32_16X16X4_F32.
// Block = 256 threads (8 waves). Wave w owns N-tile w; loops over 8 M-subtiles
// of a 128-row block. W (64 KB) staged in LDS once per block.
__global__ __launch_bounds__(256) void k_gemm(const float* __restrict__ X,
                                              const float* __restrict__ W,
                                              float* __restrict__ H, int M) {
  __shared__ float Wl[HID * HID];                 // 64 KB
  const int t = threadIdx.x;
  {
    const float4* __restrict__ Wv = (const float4*)W;
    float4* Wlv = (float4*)Wl;
#pragma unroll
    for (int i = 0; i < 16; ++i)                  // 4096 float4 / 256 threads
      Wlv[t + 256 * i] = Wv[t + 256 * i];
  }
  __syncthreads();

  const int wave = t >> 5;
  const int lane = t & 31;
  const int m16  = lane & 15;                     // A row / B,C col within tile
  const int kh   = (lane >> 4) << 1;              // K half select: 0 or 2
  const int rsel = (lane >> 4) << 3;              // C row offset: 0 or 8
  const int ncol = (wave << 4) + m16;             // global column 0..127
  const long rowBase = (long)blockIdx.x * 128;

  for (int mt = 0; mt < 8; ++mt) {
    const long row0 = rowBase + (mt << 4);
    if (row0 >= M) return;                        // N % 16 == 0: no partial tiles
    const float* __restrict__ Xr = X + (row0 + m16) * (long)HID;
    v8f acc = {};
#pragma unroll
    for (int k0 = 0; k0 < HID; k0 += 4) {
      v2f a = *(const v2f*)(Xr + k0 + kh);        // A[m][k0+kh], A[m][k0+kh+1]
      v2f b;
      b.x = Wl[(k0 + kh) * HID + ncol];           // B[k0+kh][n]
      b.y = Wl[(k0 + kh + 1) * HID + ncol];       // B[k0+kh+1][n]
      acc = __builtin_amdgcn_wmma_f32_16x16x4_f32(
          false, a, false, b, (short)0, acc, false, false);
    }
    float* __restrict__ Hp = H + row0 * (long)HID + ncol;
#pragma unroll
    for (int p = 0; p < 8; ++p)                   // C row = p + rsel
      Hp[(long)(p + rsel) * HID] = acc[p];
  }
}

// ---------------------------------------------------------------- self-loop init
// OUT[i][:] = dinv[i]^2 * H[i][:]   (one float4 per thread; fully overwrites OUT)
__global__ __launch_bounds__(256) void k_self(const float* __restrict__ H,
                                              const float* __restrict__ dinv,
                                              float* __restrict__ OUT, int n32) {
  int i = blockIdx.x * 256 + threadIdx.x;
  if (i >= n32) return;
  int node = i >> 5;
  float d = dinv[node];
  float s = d * d;
  float4 v = ((const float4*)H)[i];
  v.x *= s; v.y *= s; v.z *= s; v.w *= s;
  ((float4*)OUT)[i] = v;
}

// ---------------------------------------------------------------- edge scatter
// One wave32 per edge: lane handles 4 contiguous floats (coalesced 512B gather,
// L2-resident since feature table is 51.2MB < 192MB L2), then 4 f32 atomics.
__global__ __launch_bounds__(256) void k_edges(const float* __restrict__ H,
                                               const float* __restrict__ dinv,
                                               const int* __restrict__ src,
                                               const int* __restrict__ dst,
                                               float* __restrict__ OUT, int E) {
  int g = blockIdx.x * 256 + threadIdx.x;
  int e = g >> 5;
  if (e >= E) return;
  int lane = g & 31;
  int s = src[e];
  int d = dst[e];
  float nrm = dinv[s] * dinv[d];
  const float4 hv = *(const float4*)(H + (size_t)s * HID + (lane << 2));
  float* o = OUT + (size_t)d * HID + (lane << 2);
  atomicAdd(o + 0, hv.x * nrm);
  atomicAdd(o + 1, hv.y * nrm);
  atomicAdd(o + 2, hv.z * nrm);
  atomicAdd(o + 3, hv.w * nrm);
}

// ---------------------------------------------------------------- bias + ReLU
__global__ __launch_bounds__(256) void k_bias_relu(float* __restrict__ OUT,
                                                   const float* __restrict__ b, int n32) {
  int i = blockIdx.x * 256 + threadIdx.x;
  if (i >= n32) return;
  float4 v = ((float4*)OUT)[i];
  float4 bb = ((const float4*)b)[i & 31];
  v.x = fmaxf(v.x + bb.x, 0.0f);
  v.y = fmaxf(v.y + bb.y, 0.0f);
  v.z = fmaxf(v.z + bb.z, 0.0f);
  v.w = fmaxf(v.w + bb.w, 0.0f);
  ((float4*)OUT)[i] = v;
}

// ---------------------------------------------------------------- launcher
extern "C" void kernel_launch(void* const* d_in, const int* in_sizes, int n_in,
                              void* d_out, int out_size, void* d_ws, size_t ws_size,
                              hipStream_t stream) {
  (void)n_in; (void)out_size; (void)ws_size;
  const float* x  = (const float*)d_in[0];
  const int*   ei = (const int*)d_in[1];
  const float* W1 = (const float*)d_in[2];
  const float* b1 = (const float*)d_in[3];
  const float* W2 = (const float*)d_in[4];
  const float* b2 = (const float*)d_in[5];

  const int N = in_sizes[0] / HID;
  const int E = in_sizes[1] / 2;
  const int* src = ei;          // edge_index[0]
  const int* dst = ei + E;      // edge_index[1]

  float* H    = (float*)d_ws;             // N*128 floats (GEMM output, 51.2 MB)
  float* dinv = H + (size_t)N * HID;      // N floats
  float* OUT  = (float*)d_out;            // accumulator + final output

  const int n32 = N * 32;                 // float4 elements over N*128
  const int eth = E * 32;                 // threads for edge kernel
  const int nbN   = (N + 255) / 256;
  const int nbE   = (E + 255) / 256;
  const int nb32  = (n32 + 255) / 256;
  const int nbEth = (eth + 255) / 256;
  const int nbM   = (N + 127) / 128;

  // normalization coefficients
  k_deg_init <<<nbN, 256, 0, stream>>>(dinv, N);
  k_deg_edges<<<nbE, 256, 0, stream>>>(dinv, dst, E);
  k_rsqrt    <<<nbN, 256, 0, stream>>>(dinv, N);

  // layer 1
  k_gemm     <<<nbM, 256, 0, stream>>>(x, W1, H, N);
  k_self     <<<nb32, 256, 0, stream>>>(H, dinv, OUT, n32);
  k_edges    <<<nbEth, 256, 0, stream>>>(H, dinv, src, dst, OUT, E);
  k_bias_relu<<<nb32, 256, 0, stream>>>(OUT, b1, n32);

  // layer 2 (GEMM reads OUT, writes H; then OUT is rebuilt as accumulator)
  k_gemm     <<<nbM, 256, 0, stream>>>(OUT, W2, H, N);
  k_self     <<<nb32, 256, 0, stream>>>(H, dinv, OUT, n32);
  k_edges    <<<nbEth, 256, 0, stream>>>(H, dinv, src, dst, OUT, E);
  k_bias_relu<<<nb32, 256, 0, stream>>>(OUT, b2, n32);
}